// ContrastiveLoss_38285338477163
// MI455X (gfx1250) — compile-verified
//
#include <hip/hip_runtime.h>
#include <hip/hip_bf16.h>

typedef float v2f __attribute__((ext_vector_type(2)));
typedef float v8f __attribute__((ext_vector_type(8)));

#define TEMP_F 0.07f
#define EPS_F 1e-12f

// One wave (32 threads) per group of 4 samples: a 16x1024 f32 tile.
// Computes G = X * X^T (16x16) via 256 chained V_WMMA_F32_16X16X4_F32,
// then per-sample contrastive-loss epilogue on lanes 0..3.
__global__ __launch_bounds__(32)
void ContrastiveLoss_gram_kernel(const float* __restrict__ emb,
                                 const int* __restrict__ labels,
                                 float* __restrict__ ws)
{
    const int lane  = threadIdx.x;          // 0..31
    const int group = blockIdx.x;           // 4 samples per group

    // --- load layout: identical for WMMA A (16x4) and B (4x16 = A^T) ---
    // lane<16 : row=lane,    cols {k, k+1}
    // lane>=16: row=lane-16, cols {k+2, k+3}
    const int row     = lane & 15;
    const int colBase = (lane >> 4) * 2;
    const float* p = emb + (size_t)group * (16 * 1024) + (size_t)row * 1024 + colBase;

    v8f c = {};  // f32 accumulator, starts at 0
#pragma unroll 8
    for (int kk = 0; kk < 256; ++kk) {
        v2f ab = *(const v2f*)(p + kk * 4);   // global_load_b64, 16B stride
        // D = A*B + C ; A and B are the same registers (Gram matrix)
        c = __builtin_amdgcn_wmma_f32_16x16x4_f32(
                /*neg_a=*/false, ab, /*neg_b=*/false, ab,
                /*c_mod=*/(short)0, c, /*reuse_a=*/false, /*reuse_b=*/false);
    }

    // --- spill C/D (M = v or v+8, N = lane%16) to LDS ---
    __shared__ float g[16 * 16];
    __shared__ float partL[4];
    __shared__ float partC[4];
    const int rbase = (lane < 16) ? 0 : 8;
    const int col   = lane & 15;
#pragma unroll
    for (int v = 0; v < 8; ++v)
        g[(rbase + v) * 16 + col] = c[v];
    __syncthreads();

    // --- per-sample epilogue: lanes 0..3 handle samples group*4+s ---
    if (lane < 4) {
        const int s         = lane;
        const int sampleIdx = group * 4 + s;

        float G[4][4];
#pragma unroll
        for (int i = 0; i < 4; ++i)
#pragma unroll
            for (int j = 0; j < 4; ++j)
                G[i][j] = g[(4 * s + i) * 16 + (4 * s + j)];

        float d[4];
#pragma unroll
        for (int i = 0; i < 4; ++i)
            d[i] = fmaxf(sqrtf(G[i][i]), EPS_F);   // max(||x||, eps)

        bool pos[4], neg[4];
        int P = 0, N = 0;
#pragma unroll
        for (int k = 0; k < 4; ++k) {
            const int lab = labels[sampleIdx * 4 + k];
            pos[k] = (lab == 1);
            neg[k] = (lab == 0);
            P += pos[k];
            N += neg[k];
        }
        const bool valid = (P > 0) && (N > 0);

        float loss = 0.0f, posSum = 0.0f;
#pragma unroll
        for (int i = 0; i < 4; ++i) {
            if (pos[i]) {
                float negSum = 0.0f;
#pragma unroll
                for (int j = 0; j < 4; ++j) {
                    const float sim = G[i][j] / (d[i] * d[j]) / TEMP_F;
                    const float e   = expf(sim);
                    if (pos[j]) posSum += e;
                    if (neg[j]) negSum += e;
                }
                if (valid) loss += log1pf(negSum);
            }
        }
        if (valid && P > 1)
            loss -= logf(posSum / (float)(P * P));   // -log(pos_mean)

        partL[s] = loss;
        partC[s] = valid ? (float)(P + ((P > 1) ? 1 : 0)) : 0.0f;
    }
    __syncthreads();

    if (lane == 0) {
        ws[2 * group + 0] = partL[0] + partL[1] + partL[2] + partL[3];
        ws[2 * group + 1] = partC[0] + partC[1] + partC[2] + partC[3];
    }
}

// Deterministic final reduction of per-block (loss, count) partials.
__global__ __launch_bounds__(256)
void ContrastiveLoss_reduce_kernel(const float* __restrict__ ws,
                                   float* __restrict__ out, int n)
{
    __shared__ float st[256];
    __shared__ float sc[256];
    float t = 0.0f, c0 = 0.0f;
    for (int i = threadIdx.x; i < n; i += 256) {
        t  += ws[2 * i + 0];
        c0 += ws[2 * i + 1];
    }
    st[threadIdx.x] = t;
    sc[threadIdx.x] = c0;
    __syncthreads();
    for (int off = 128; off > 0; off >>= 1) {
        if ((int)threadIdx.x < off) {
            st[threadIdx.x] += st[threadIdx.x + off];
            sc[threadIdx.x] += sc[threadIdx.x + off];
        }
        __syncthreads();
    }
    if (threadIdx.x == 0)
        out[0] = st[0] / fmaxf(sc[0], 1.0f);
}

extern "C" void kernel_launch(void* const* d_in, const int* in_sizes, int n_in,
                              void* d_out, int out_size, void* d_ws, size_t ws_size,
                              hipStream_t stream)
{
    const float* emb    = (const float*)d_in[0];   // (B, 4, 1024) f32
    const int*   labels = (const int*)d_in[1];     // (B, 4) int
    float* out = (float*)d_out;
    float* ws  = (float*)d_ws;

    const int B      = in_sizes[1] / 4;   // 65536
    const int groups = B / 4;             // 16384 groups of 4 samples

    ContrastiveLoss_gram_kernel<<<groups, 32, 0, stream>>>(emb, labels, ws);
    ContrastiveLoss_reduce_kernel<<<1, 256, 0, stream>>>(ws, out, groups);
}